// ShortConvolution_41575283425757
// MI455X (gfx1250) — compile-verified
//
#include <hip/hip_runtime.h>

// Causal depthwise conv1d, dim-last: y[b,t,d] = sum_{j<4} w[d,j] * x[b, t-3+j, d]
// x: (4, 4096, 2048) f32, w: (2048, 4) f32, y: (4, 4096, 2048) f32.
// Memory-bound (~1 FLOP/byte, ~268 MB total => ~11.5us floor @ 23.3 TB/s).
// Design: one TDM (tensor_load_to_lds) descriptor stages each (35 x 256f) x-tile
// into LDS; compute is a register sliding window with coalesced b128 stores.

#define B_N 4
#define L_N 4096
#define D_N 2048
#define K_N 4

typedef __attribute__((address_space(3))) float lds_f;
typedef unsigned int v4u __attribute__((ext_vector_type(4)));
typedef unsigned int v8u __attribute__((ext_vector_type(8)));

constexpr int D_TILE   = 256;              // channels per block (64 float4)
constexpr int T_TILE   = 32;               // output timesteps per block
constexpr int ROWS     = T_TILE + K_N - 1; // 35 staged rows (3 halo)
constexpr int NTHREADS = 256;              // 8 waves (wave32)

__global__ __launch_bounds__(NTHREADS)
void ShortConvolution_41575283425757_kernel(const float* __restrict__ x,
                                            const float* __restrict__ w,
                                            float* __restrict__ y) {
    __shared__ float smem[ROWS * D_TILE];  // 35,840 B

    const int tid = threadIdx.x;
    const int b   = blockIdx.z;
    const int t0  = blockIdx.y * T_TILE;
    const int d0  = blockIdx.x * D_TILE;

    const float* xb = x + (size_t)b * L_N * D_N;

    // ---- Causal zero padding: first time-tile needs 3 zero halo rows ----
    const int row_lo = (t0 == 0) ? (K_N - 1) : 0;   // rows [0,row_lo) are zeros
    if (t0 == 0 && tid < (K_N - 1) * (D_TILE / 4)) {
        float4 z = make_float4(0.f, 0.f, 0.f, 0.f);
        *(float4*)&smem[tid * 4] = z;
    }

    // ---- Stage x tile with one Tensor Data Mover descriptor (wave 0) ----
    if (tid < 32) {
        const int rows = ROWS - row_lo;  // 32 or 35 valid rows
        const unsigned lds_addr =
            (unsigned)(unsigned long long)(lds_f*)&smem[row_lo * D_TILE];
        const unsigned long long gaddr =
            (unsigned long long)(xb + (size_t)(t0 - (K_N - 1) + row_lo) * D_N + d0);

        // D# group 0: count=1 | lds_addr | global_addr[56:0] | type=2
        v4u g0;
        g0.x = 0x1u;
        g0.y = lds_addr;
        g0.z = (unsigned)gaddr;
        g0.w = ((unsigned)(gaddr >> 32) & 0x01FFFFFFu) | 0x80000000u;

        // D# group 1: data_size=2 (4B); tensor_dim0=256; tensor_dim1=rows;
        // tile_dim0=256; tile_dim1=rows; tensor_dim0_stride=2048 (D_N).
        v8u g1;
        g1.s0 = 0x2u << 16;                     // data_size=4B, mask=0, no pad/iter
        g1.s1 = (unsigned)D_TILE << 16;         // tensor_dim0[15:0]
        g1.s2 = (unsigned)rows << 16;           // tensor_dim0[31:16]=0 | tensor_dim1[15:0]
        g1.s3 = (unsigned)D_TILE << 16;         // tensor_dim1[31:16]=0 | tile_dim0
        g1.s4 = (unsigned)rows;                 // tile_dim1 | tile_dim2=0
        g1.s5 = (unsigned)D_N;                  // tensor_dim0_stride[31:0]
        g1.s6 = 0u;                             // stride0[47:32] | stride1[15:0]
        g1.s7 = 0u;                             // tensor_dim1_stride[47:16]

        v4u g2 = {0u, 0u, 0u, 0u};              // dims >= 2 unused
        v4u g3 = {0u, 0u, 0u, 0u};

        asm volatile("tensor_load_to_lds %0, %1, %2, %3"
                     :: "s"(g0), "s"(g1), "s"(g2), "s"(g3)
                     : "memory");
        __builtin_amdgcn_s_wait_tensorcnt(0);   // TDM done -> LDS data valid
    }
    __syncthreads();

    // ---- Compute: each thread owns 4 channels x 8 timesteps ----
    const int dchunk = tid & 63;      // float4 chunk within D_TILE
    const int tg     = tid >> 6;      // 0..3
    const int tbase  = tg * 8;        // local output timestep start
    const int dglob  = d0 + dchunk * 4;

    // w is (D, 4) row-major: taps for one channel are a contiguous float4.
    const float4 w0 = *(const float4*)(w + (size_t)(dglob + 0) * K_N);
    const float4 w1 = *(const float4*)(w + (size_t)(dglob + 1) * K_N);
    const float4 w2 = *(const float4*)(w + (size_t)(dglob + 2) * K_N);
    const float4 w3 = *(const float4*)(w + (size_t)(dglob + 3) * K_N);

    const int scol = dchunk * 4;
    // register sliding window of 4 staged rows
    float4 r0 = *(const float4*)&smem[(tbase + 0) * D_TILE + scol];
    float4 r1 = *(const float4*)&smem[(tbase + 1) * D_TILE + scol];
    float4 r2 = *(const float4*)&smem[(tbase + 2) * D_TILE + scol];
    float4 r3 = *(const float4*)&smem[(tbase + 3) * D_TILE + scol];

    float* yb = y + ((size_t)b * L_N + (t0 + tbase)) * D_N + dglob;

#pragma unroll
    for (int i = 0; i < 8; ++i) {
        float4 acc;
        // tap order j = 0..3 matches the reference accumulation order
        acc.x = w0.x * r0.x + w0.y * r1.x + w0.z * r2.x + w0.w * r3.x;
        acc.y = w1.x * r0.y + w1.y * r1.y + w1.z * r2.y + w1.w * r3.y;
        acc.z = w2.x * r0.z + w2.y * r1.z + w2.z * r2.z + w2.w * r3.z;
        acc.w = w3.x * r0.w + w3.y * r1.w + w3.z * r2.w + w3.w * r3.w;
        *(float4*)(yb + (size_t)i * D_N) = acc;
        r0 = r1; r1 = r2; r2 = r3;
        if (i < 7)
            r3 = *(const float4*)&smem[(tbase + i + 4) * D_TILE + scol];
    }
}

extern "C" void kernel_launch(void* const* d_in, const int* in_sizes, int n_in,
                              void* d_out, int out_size, void* d_ws, size_t ws_size,
                              hipStream_t stream) {
    (void)in_sizes; (void)n_in; (void)out_size; (void)d_ws; (void)ws_size;
    const float* x = (const float*)d_in[0];
    const float* w = (const float*)d_in[1];
    float*       y = (float*)d_out;

    dim3 grid(D_N / D_TILE, L_N / T_TILE, B_N);  // (8, 128, 4)
    ShortConvolution_41575283425757_kernel<<<grid, NTHREADS, 0, stream>>>(x, w, y);
}